// DynamicPillarFeatureNet_78537771975385
// MI455X (gfx1250) — compile-verified
//
#include <hip/hip_runtime.h>

typedef __attribute__((ext_vector_type(2))) float v2f;
typedef __attribute__((ext_vector_type(8))) float v8f;

// ---------------------------------------------------------------------------
// Pass 1: per-voxel xyz sums + counts via float atomics (L2-resident, 3.2MB)
// ws layout: [voxel*4 + {0,1,2}] = sum(xyz), [voxel*4 + 3] = count
// ---------------------------------------------------------------------------
__global__ __launch_bounds__(256) void scatter_sum_kernel(
    const float* __restrict__ features, const int* __restrict__ p2v,
    float* __restrict__ ws, int n_points)
{
    int i = blockIdx.x * blockDim.x + threadIdx.x;
    if (i >= n_points) return;
    float4 f = ((const float4*)features)[i];
    unsigned v = (unsigned)p2v[i];
    atomicAdd(ws + (size_t)v * 4 + 0, f.x);
    atomicAdd(ws + (size_t)v * 4 + 1, f.y);
    atomicAdd(ws + (size_t)v * 4 + 2, f.z);
    atomicAdd(ws + (size_t)v * 4 + 3, 1.0f);
}

// ---------------------------------------------------------------------------
// Pass 2: mean = sum / max(cnt, 1)   (in place, sums -> means)
// ---------------------------------------------------------------------------
__global__ __launch_bounds__(256) void finalize_mean_kernel(
    float* __restrict__ ws, int n_voxels)
{
    int v = blockIdx.x * blockDim.x + threadIdx.x;
    if (v >= n_voxels) return;
    float c = fmaxf(ws[(size_t)v * 4 + 3], 1.0f);
    ws[(size_t)v * 4 + 0] = ws[(size_t)v * 4 + 0] / c;
    ws[(size_t)v * 4 + 1] = ws[(size_t)v * 4 + 1] / c;
    ws[(size_t)v * 4 + 2] = ws[(size_t)v * 4 + 2] / c;
}

// ---------------------------------------------------------------------------
// Pass 3: fused feature-build + 7x64 GEMM via V_WMMA_F32_16X16X4_F32 + BN +
// ReLU + segment-max scatter.  BN scale folded into B; BN bias folded into
// the K=7 padding row (A[:,7] = 1, B[7][n] = bias[n]) so the C operand is
// inline zero.  ReLU + segment-max are fused into a single int-bit atomicMax
// against a zeroed output: non-negative floats order as positive ints, and
// any negative float's bit pattern is a negative int that can never beat the
// zero init -- so no per-element VALU is needed at all.  Wave id forced
// scalar via readfirstlane so tile guards are s_cmp/s_cbranch (EXEC stays
// all-1s for WMMA).  Atomics use SADDR + u32 voffset with nt*64B folded into
// the immediate offset.
// ---------------------------------------------------------------------------
__global__ __launch_bounds__(256) void pfn_wmma_kernel(
    const float* __restrict__ features,   // N x 4
    const float* __restrict__ W,          // 7 x 64
    const float* __restrict__ bn_gamma,   // 64
    const float* __restrict__ bn_beta,    // 64
    const float* __restrict__ bn_mean,    // 64
    const float* __restrict__ bn_var,     // 64
    const int*   __restrict__ p2v,        // N
    const int*   __restrict__ coors,      // M x 4
    const float* __restrict__ vmean,      // M x 4 (xyz means + cnt)
    float*       __restrict__ out,        // M x 64 (pre-zeroed)
    int n_points)
{
    const int lane = threadIdx.x & 31;
    // Wave id is uniform within the wave: force it into an SGPR so all tile
    // guards become scalar branches.
    const int wave = __builtin_amdgcn_readfirstlane(threadIdx.x >> 5);
    const int tile = blockIdx.x * (blockDim.x >> 5) + wave;
    const int n_tiles = (n_points + 15) >> 4;
    if (tile >= n_tiles) return;           // scalar branch

    const int m     = lane & 15;           // point-in-tile (A rows / D cols)
    const int khalf = lane >> 4;           // lane half selects K pairs
    const int t16   = tile * 16;

    // ---- B-matrix with BN scale folded in; bias lives in row K=7 ----------
    // B layout (4x16 f32): lanes 0-15 hold rows {K,K+1} in {v0,v1} for the
    // low K pair; lanes 16-31 hold rows {K+2,K+3}.
    v2f Bm[4][2];
#pragma unroll
    for (int nt = 0; nt < 4; ++nt) {
        int n = nt * 16 + m;
        float sc = rsqrtf(bn_var[n] + 1e-3f) * bn_gamma[n];
        int k0 = khalf * 2;        // 0 or 2
        Bm[nt][0].x = W[(k0 + 0) * 64 + n] * sc;
        Bm[nt][0].y = W[(k0 + 1) * 64 + n] * sc;
        if (khalf == 0) {          // rows 4,5
            Bm[nt][1].x = W[4 * 64 + n] * sc;
            Bm[nt][1].y = W[5 * 64 + n] * sc;
        } else {                   // rows 6,7 : row7 = BN bias
            Bm[nt][1].x = W[6 * 64 + n] * sc;
            Bm[nt][1].y = bn_beta[n] - bn_mean[n] * sc;
        }
    }

    // ---- Per-point feature construction (each lane: point m of the tile) --
    int p  = t16 + m;
    int pc = (p < n_points) ? p : (n_points - 1);
    float4 f4 = ((const float4*)features)[pc];
    unsigned v = (unsigned)p2v[pc];
    int4 cr = ((const int4*)coors)[v];
    float mx = vmean[(size_t)v * 4 + 0];
    float my = vmean[(size_t)v * 4 + 1];
    float mz = vmean[(size_t)v * 4 + 2];
    // centers = (coors[:,[3,2,1]] + 0.5) * (0.2,0.2,4.0) + (0,-40,-3)
    float cx = ((float)cr.w + 0.5f) * 0.2f + 0.0f;
    float cy = ((float)cr.z + 0.5f) * 0.2f - 40.0f;
    float cz = ((float)cr.y + 0.5f) * 4.0f - 3.0f;
    float cl_x = f4.x - mx, cl_y = f4.y - my, cl_z = f4.z - mz;
    float ce_x = f4.x - cx, ce_y = f4.y - cy, ce_z = f4.z - cz;

    // A layout (16x4 f32): lanes 0-15: v0=K0,v1=K1 ; lanes 16-31: v0=K2,v1=K3
    // feats = [cl_x cl_y cl_z ce_x ce_y ce_z w 1]; the trailing 1 activates
    // the bias row of B.
    v2f A0, A1;
    if (khalf == 0) { A0.x = cl_x; A0.y = cl_y; A1.x = ce_y; A1.y = ce_z; }
    else            { A0.x = cl_z; A0.y = ce_x; A1.x = f4.w; A1.y = 1.0f; }

    // ---- 8 WMMAs: 4 independent two-deep chains, C starts at inline 0 -----
    v8f acc[4];
#pragma unroll
    for (int nt = 0; nt < 4; ++nt) {
        v8f c = {};                            // -> inline 0 SRC2
        c = __builtin_amdgcn_wmma_f32_16x16x4_f32(
            false, A0, false, Bm[nt][0], (short)0, c, false, false);
        c = __builtin_amdgcn_wmma_f32_16x16x4_f32(
            false, A1, false, Bm[nt][1], (short)0, c, false, false);
        acc[nt] = c;
    }

    // ---- Row voxel ids: redundant same-cacheline loads (no shuffles) ------
    // Lane-half h needs rows h*8 .. h*8+7; all 16 lanes of a half load the
    // same address -> one cacheline broadcast per load, single wait window.
    const int row0 = khalf << 3;
    unsigned ofs[8];                           // u32 element offsets into out
#pragma unroll
    for (int r = 0; r < 8; ++r) {
        int rp = t16 + row0 + r;
        int rpc = (rp < n_points) ? rp : (n_points - 1);
        unsigned vr = (unsigned)p2v[rpc];
        ofs[r] = vr * 64u + (unsigned)m;       // < 12.8M, fits u32
    }

    int* outi = (int*)out;
    if (t16 + 16 <= n_points) {                // scalar branch (t16 uniform)
        // Fast path (all tiles when N % 16 == 0): no exec masking, and the
        // raw accumulator bits go straight to the atomic (ReLU fused in).
#pragma unroll
        for (int nt = 0; nt < 4; ++nt) {
#pragma unroll
            for (int r = 0; r < 8; ++r) {
                atomicMax(outi + (size_t)ofs[r] + nt * 16,  // nt*64B -> ioffset
                          __float_as_int(acc[nt][r]));
            }
        }
    } else {
        // Tail tile: guard rows beyond n_points (rare; at most one wave).
#pragma unroll
        for (int nt = 0; nt < 4; ++nt) {
#pragma unroll
            for (int r = 0; r < 8; ++r) {
                if (t16 + row0 + r < n_points) {
                    atomicMax(outi + (size_t)ofs[r] + nt * 16,
                              __float_as_int(acc[nt][r]));
                }
            }
        }
    }
}

extern "C" void kernel_launch(void* const* d_in, const int* in_sizes, int n_in,
                              void* d_out, int out_size, void* d_ws, size_t ws_size,
                              hipStream_t stream) {
    const float* features = (const float*)d_in[0];   // N x 4
    const float* W        = (const float*)d_in[1];   // 7 x 64
    const float* bn_gamma = (const float*)d_in[2];
    const float* bn_beta  = (const float*)d_in[3];
    const float* bn_mean  = (const float*)d_in[4];
    const float* bn_var   = (const float*)d_in[5];
    const int*   p2v      = (const int*)d_in[6];     // N
    const int*   coors    = (const int*)d_in[7];     // M x 4

    const int n_points = in_sizes[0] / 4;
    const int n_voxels = in_sizes[7] / 4;

    float* out = (float*)d_out;                       // [M*64 f32][M*4 i32]
    float* ws  = (float*)d_ws;                        // M x 4 f32 sums/cnt

    // Deterministic init each call (graph-capture safe stream ops only).
    hipMemsetAsync(ws,  0, (size_t)n_voxels * 4  * sizeof(float), stream);
    hipMemsetAsync(out, 0, (size_t)n_voxels * 64 * sizeof(float), stream);
    hipMemcpyAsync(out + (size_t)n_voxels * 64, coors,
                   (size_t)n_voxels * 4 * sizeof(int),
                   hipMemcpyDeviceToDevice, stream);

    scatter_sum_kernel<<<(n_points + 255) / 256, 256, 0, stream>>>(
        features, p2v, ws, n_points);
    finalize_mean_kernel<<<(n_voxels + 255) / 256, 256, 0, stream>>>(
        ws, n_voxels);

    const int n_tiles = (n_points + 15) / 16;         // one wave32 per tile
    const int waves_per_block = 256 / 32;
    const int blocks = (n_tiles + waves_per_block - 1) / waves_per_block;
    pfn_wmma_kernel<<<blocks, 256, 0, stream>>>(
        features, W, bn_gamma, bn_beta, bn_mean, bn_var,
        p2v, coors, ws, out, n_points);
}